// NIQE_51831665328157
// MI455X (gfx1250) — compile-verified
//
#include <hip/hip_runtime.h>
#include <hip/hip_bf16.h>
#include <math.h>

typedef __attribute__((ext_vector_type(2))) float v2f;
typedef __attribute__((ext_vector_type(8))) float v8f;

// ---------------------------------------------------------------------------
// R_GAM table: exp(2*lgamma(2/g) - lgamma(1/g) - lgamma(3/g)), g = 0.2..10 step 1e-3
// ---------------------------------------------------------------------------
__global__ void rgam_kernel(float* __restrict__ rgam) {
    int k = blockIdx.x * blockDim.x + threadIdx.x;
    if (k >= 9801) return;
    double g = 0.2 + 0.001 * (double)k;
    double r = exp(2.0 * lgamma(2.0 / g) - lgamma(1.0 / g) - lgamma(3.0 / g));
    rgam[k] = (float)r;
}

// ---------------------------------------------------------------------------
// MSCN: mu = G*img, ex2 = G*img^2 (edge pad), out = (img-mu)/(sqrt(|ex2-mu^2|)+1)
// 16x16 output tile per workgroup, 22x22 LDS tile (halo 3).
// ---------------------------------------------------------------------------
__global__ __launch_bounds__(256) void mscn_kernel(const float* __restrict__ img,
                                                   const float* __restrict__ gw,
                                                   float* __restrict__ out,
                                                   int H, int W) {
    __shared__ float g[49];
    __shared__ float tile[22][22];
    const int tx = threadIdx.x, ty = threadIdx.y;
    const int tid = ty * 16 + tx;
    if (tid < 49) g[tid] = gw[tid];
    const int b = blockIdx.z;
    const int x0 = blockIdx.x * 16, y0 = blockIdx.y * 16;
    const float* im = img + (size_t)b * H * W;
    for (int i = tid; i < 22 * 22; i += 256) {
        int ry = i / 22, rx = i % 22;
        int yy = min(max(y0 + ry - 3, 0), H - 1);
        int xx = min(max(x0 + rx - 3, 0), W - 1);
        tile[ry][rx] = im[(size_t)yy * W + xx];
    }
    __syncthreads();
    float mu = 0.f, ex2 = 0.f;
#pragma unroll
    for (int dy = 0; dy < 7; ++dy)
#pragma unroll
        for (int dx = 0; dx < 7; ++dx) {
            float p = tile[ty + dy][tx + dx];
            float w = g[dy * 7 + dx];
            mu  = fmaf(w, p, mu);
            ex2 = fmaf(w, p * p, ex2);
        }
    float c = tile[ty + 3][tx + 3];
    float sigma = sqrtf(fabsf(ex2 - mu * mu));
    out[(size_t)b * H * W + (size_t)(y0 + ty) * W + (x0 + tx)] = (c - mu) / (sigma + 1.0f);
}

// ---------------------------------------------------------------------------
// Bicubic half-resize: fixed 8-tap filter (weights position independent, sum=1),
// symmetric-reflect index map k' = (k<0 ? -k-1 : k>=n ? 2n-1-k : k), k = 2y-4+j.
// ---------------------------------------------------------------------------
__device__ __forceinline__ int reflect_idx(int k, int n) {
    if (k < 0) k = -k - 1;
    if (k >= n) k = 2 * n - 1 - k;
    return k;
}

__device__ __constant__ float RW[10] = {
    0.f, -0.01171875f, -0.03515625f, 0.11328125f, 0.43359375f,
    0.43359375f, 0.11328125f, -0.03515625f, -0.01171875f, 0.f};

__global__ __launch_bounds__(256) void resize_h_kernel(const float* __restrict__ in,
                                                       float* __restrict__ out,
                                                       int Hin, int Hout, int W) {
    int x = blockIdx.x * blockDim.x + threadIdx.x;
    int y = blockIdx.y, b = blockIdx.z;
    if (x >= W) return;
    const float* im = in + (size_t)b * Hin * W;
    float acc = 0.f;
#pragma unroll
    for (int j = 1; j <= 8; ++j) {
        int k = reflect_idx(2 * y - 4 + j, Hin);
        acc = fmaf(RW[j], im[(size_t)k * W + x], acc);
    }
    out[(size_t)b * Hout * W + (size_t)y * W + x] = acc;
}

__global__ __launch_bounds__(256) void resize_w_kernel(const float* __restrict__ in,
                                                       float* __restrict__ out,
                                                       int Win, int Wout, int H) {
    int x = blockIdx.x * blockDim.x + threadIdx.x;
    int y = blockIdx.y, b = blockIdx.z;
    if (x >= Wout) return;
    const float* row = in + ((size_t)b * H + y) * Win;
    float acc = 0.f;
#pragma unroll
    for (int j = 1; j <= 8; ++j) {
        int k = reflect_idx(2 * x - 4 + j, Win);
        acc = fmaf(RW[j], row[k], acc);
    }
    out[((size_t)b * H + y) * Wout + x] = acc;
}

// ---------------------------------------------------------------------------
// AGGD features. grid=(variant 5, blk 192, batch 8), 256 threads.
// v=0: raw block; v=1..4: block * roll(block, sh) with wrap inside the block.
// ---------------------------------------------------------------------------
__global__ __launch_bounds__(256) void feature_kernel(const float* __restrict__ norm,
                                                      const float* __restrict__ rgam,
                                                      float* __restrict__ feats,
                                                      int H, int W, int B, int featOff) {
    const int v = blockIdx.x;
    const int blk = blockIdx.y;
    const int b = blockIdx.z;
    const int nh = 12;
    const int ih = blk % nh, iw = blk / nh;   // blockify: blk = iw*nh + ih
    const float* base = norm + (size_t)b * H * W + (size_t)(ih * B) * W + (size_t)(iw * B);
    int sh0 = 0, sh1 = 0;
    if (v == 1) { sh0 = 0; sh1 = 1; }
    else if (v == 2) { sh0 = 1; sh1 = 0; }
    else if (v == 3) { sh0 = 1; sh1 = 1; }
    else if (v == 4) { sh0 = 1; sh1 = -1; }
    const int tid = threadIdx.x;
    const int N = B * B;

    float cl = 0.f, cr = 0.f, sl2 = 0.f, sr2 = 0.f, sab = 0.f, ssq = 0.f;
    for (int idx = tid; idx < N; idx += 256) {
        int i = idx / B, j = idx % B;
        float x = base[(size_t)i * W + j];
        float val;
        if (v == 0) {
            val = x;
        } else {
            int i2 = i - sh0; if (i2 < 0) i2 += B;
            int j2 = j - sh1; if (j2 < 0) j2 += B; else if (j2 >= B) j2 -= B;
            val = x * base[(size_t)i2 * W + j2];
        }
        float v2 = val * val;
        if (val < 0.f) { cl += 1.f; sl2 += v2; }
        else if (val > 0.f) { cr += 1.f; sr2 += v2; }
        sab += fabsf(val);
        ssq += v2;
    }

    __shared__ float red[6][256];
    red[0][tid] = cl; red[1][tid] = cr; red[2][tid] = sl2;
    red[3][tid] = sr2; red[4][tid] = sab; red[5][tid] = ssq;
    __syncthreads();
    for (int s = 128; s > 0; s >>= 1) {
        if (tid < s)
            for (int q = 0; q < 6; ++q) red[q][tid] += red[q][tid + s];
        __syncthreads();
    }
    const float CL = red[0][0], CR = red[1][0], SL2 = red[2][0];
    const float SR2 = red[3][0], SAB = red[4][0], SSQ = red[5][0];

    float ls = sqrtf(SL2 / CL);
    float rs = sqrtf(SR2 / CR);
    float gh = ls / rs;
    float m1 = SAB / (float)N, m2 = SSQ / (float)N;
    float rhat = m1 * m1 / m2;
    float gh2 = gh * gh;
    float rhatnorm = rhat * (gh2 * gh + 1.f) * (gh + 1.f) / ((gh2 + 1.f) * (gh2 + 1.f));

    // parallel argmin over R_GAM (first-min-index semantics)
    float best = 3.4e38f; int bpos = 0x7fffffff;
    for (int k = tid; k < 9801; k += 256) {
        float d = fabsf(rgam[k] - rhatnorm);
        if (d < best || (d == best && k < bpos)) { best = d; bpos = k; }
    }
    __shared__ float bv[256];
    __shared__ int bi[256];
    bv[tid] = best; bi[tid] = bpos;
    __syncthreads();
    for (int s = 128; s > 0; s >>= 1) {
        if (tid < s) {
            if (bv[tid + s] < bv[tid] ||
                (bv[tid + s] == bv[tid] && bi[tid + s] < bi[tid])) {
                bv[tid] = bv[tid + s]; bi[tid] = bi[tid + s];
            }
        }
        __syncthreads();
    }
    if (tid == 0) {
        int pos = (bi[0] == 0x7fffffff) ? 0 : bi[0];
        float alpha = 0.2f + 0.001f * (float)pos;
        float cfac = sqrtf(expf(lgammaf(1.f / alpha) - lgammaf(3.f / alpha)));
        float bl = ls * cfac, br = rs * cfac;
        float* fo = feats + ((size_t)b * 192 + blk) * 36 + featOff;
        if (v == 0) {
            fo[0] = alpha;
            fo[1] = 0.5f * (bl + br);
        } else {
            float mean = (br - bl) * expf(lgammaf(2.f / alpha) - lgammaf(1.f / alpha));
            int o = 2 + 4 * (v - 1);
            fo[o] = alpha; fo[o + 1] = mean; fo[o + 2] = bl; fo[o + 3] = br;
        }
    }
}

// ---------------------------------------------------------------------------
// Per-image statistics: nanmean, centered Gram via V_WMMA_F32_16X16X4_F32
// (fp32 in / fp32 accumulate == reference einsum), SPD Cholesky solve of
// (cov_pris + cov_d)/2, quality = sqrt(diff^T M^{-1} diff). 1 wave32 / image.
// ---------------------------------------------------------------------------
__global__ __launch_bounds__(32) void stats_kernel(const float* __restrict__ feats,
                                                   const float* __restrict__ mu_pris,
                                                   const float* __restrict__ cov_pris,
                                                   float* __restrict__ out) {
    __shared__ float dmat[192][48];   // centered features, padded 36->48
    __shared__ float cov[48][48];     // Gram result
    __shared__ float mu_d[36];
    __shared__ float Mm[36][36];
    const int b = blockIdx.x;
    const int lane = threadIdx.x;

    for (int idx = lane; idx < 192 * 48; idx += 32) {
        int n = idx / 48, f = idx % 48;
        dmat[n][f] = (f < 36) ? feats[((size_t)b * 192 + n) * 36 + f] : 0.f;
    }
    __syncthreads();

    for (int f = lane; f < 36; f += 32) {
        float s = 0.f, cnt = 0.f;
        for (int n = 0; n < 192; ++n) {
            float v = dmat[n][f];
            if (!(v != v)) { s += v; cnt += 1.f; }
        }
        mu_d[f] = s / cnt;            // nanmean (NaN if all-NaN)
        float mean0 = s / 192.f;      // mean of d0 (NaN->0) over all blocks
        for (int n = 0; n < 192; ++n) {
            float v = dmat[n][f];
            if (v != v) v = 0.f;
            dmat[n][f] = v - mean0;
        }
    }
    __syncthreads();

    // cov = dmat^T * dmat / 191 via 9 tiles of 16x16, K-steps of 4.
    // A layout: A[M=lane%16, K=2*(lane/16)+r]; C layout: M = r + 8*(lane/16), N = lane%16.
    const int lhi = lane >> 4;
    const int llo = lane & 15;
    for (int ti = 0; ti < 3; ++ti) {
        for (int tj = 0; tj < 3; ++tj) {
            v8f acc = {0.f, 0.f, 0.f, 0.f, 0.f, 0.f, 0.f, 0.f};
            for (int k0 = 0; k0 < 192; k0 += 4) {
                v2f a, bb;
                a.x  = dmat[k0 + 2 * lhi + 0][ti * 16 + llo];
                a.y  = dmat[k0 + 2 * lhi + 1][ti * 16 + llo];
                bb.x = dmat[k0 + 2 * lhi + 0][tj * 16 + llo];
                bb.y = dmat[k0 + 2 * lhi + 1][tj * 16 + llo];
                acc = __builtin_amdgcn_wmma_f32_16x16x4_f32(
                    /*neg_a=*/false, a, /*neg_b=*/false, bb,
                    /*c_mod=*/(short)0, acc, /*reuse_a=*/false, /*reuse_b=*/false);
            }
#pragma unroll
            for (int r = 0; r < 8; ++r)
                cov[ti * 16 + r + 8 * lhi][tj * 16 + llo] = acc[r] * (1.0f / 191.0f);
        }
    }
    __syncthreads();

    for (int idx = lane; idx < 36 * 36; idx += 32) {
        int f = idx / 36, g = idx % 36;
        Mm[f][g] = 0.5f * (cov_pris[idx] + cov[f][g]);
    }
    __syncthreads();

    if (lane == 0) {
        // Cholesky (SPD): Mm = L L^T in-place (lower)
        for (int i = 0; i < 36; ++i) {
            for (int j = 0; j <= i; ++j) {
                float s = Mm[i][j];
                for (int k = 0; k < j; ++k) s -= Mm[i][k] * Mm[j][k];
                Mm[i][j] = (i == j) ? sqrtf(s) : s / Mm[j][j];
            }
        }
        float z[36], y[36], diff[36];
        for (int i = 0; i < 36; ++i) diff[i] = mu_pris[i] - mu_d[i];
        for (int i = 0; i < 36; ++i) {
            float s = diff[i];
            for (int k = 0; k < i; ++k) s -= Mm[i][k] * z[k];
            z[i] = s / Mm[i][i];
        }
        float q = 0.f;
        for (int i = 35; i >= 0; --i) {
            float s = z[i];
            for (int k = i + 1; k < 36; ++k) s -= Mm[k][i] * y[k];
            y[i] = s / Mm[i][i];
            q += diff[i] * y[i];
        }
        out[b] = sqrtf(q);
    }
}

// ---------------------------------------------------------------------------
extern "C" void kernel_launch(void* const* d_in, const int* in_sizes, int n_in,
                              void* d_out, int out_size, void* d_ws, size_t ws_size,
                              hipStream_t stream) {
    const float* X        = (const float*)d_in[0];   // (8,1,1152,1536)
    const float* mu_pris  = (const float*)d_in[1];   // (36,)
    const float* cov_pris = (const float*)d_in[2];   // (36,36)
    const float* gw       = (const float*)d_in[3];   // (1,1,7,7)
    float* out = (float*)d_out;                      // (8,)

    const int Bn = 8, H1 = 1152, W1 = 1536, H2 = 576, W2 = 768;

    float* bufA    = (float*)d_ws;                            // norm1 / resize tmp (aliased)
    float* imgHalf = bufA + (size_t)Bn * H1 * W1;
    float* norm2   = imgHalf + (size_t)Bn * H2 * W2;
    float* feats   = norm2 + (size_t)Bn * H2 * W2;
    float* rgam    = feats + (size_t)Bn * 192 * 36;
    float* tmp     = bufA;                                    // 8x576x1536 fits in bufA

    rgam_kernel<<<dim3((9801 + 255) / 256), dim3(256), 0, stream>>>(rgam);

    // scale 1
    mscn_kernel<<<dim3(W1 / 16, H1 / 16, Bn), dim3(16, 16), 0, stream>>>(X, gw, bufA, H1, W1);
    feature_kernel<<<dim3(5, 192, Bn), dim3(256), 0, stream>>>(bufA, rgam, feats, H1, W1, 96, 0);

    // half-resize (linear, so resize(img/255)*255 == resize(img)); tmp aliases bufA
    resize_h_kernel<<<dim3(W1 / 256, H2, Bn), dim3(256), 0, stream>>>(X, tmp, H1, H2, W1);
    resize_w_kernel<<<dim3(W2 / 256, H2, Bn), dim3(256), 0, stream>>>(tmp, imgHalf, W1, W2, H2);

    // scale 2
    mscn_kernel<<<dim3(W2 / 16, H2 / 16, Bn), dim3(16, 16), 0, stream>>>(imgHalf, gw, norm2, H2, W2);
    feature_kernel<<<dim3(5, 192, Bn), dim3(256), 0, stream>>>(norm2, rgam, feats, H2, W2, 48, 18);

    // per-image covariance (WMMA f32) + Cholesky solve
    stats_kernel<<<dim3(Bn), dim3(32), 0, stream>>>(feats, mu_pris, cov_pris, out);
}